// RecurrentMambaCell_72773925863591
// MI455X (gfx1250) — compile-verified
//
#include <hip/hip_runtime.h>
#include <hip/hip_bf16.h>

// ---------------------------------------------------------------------------
// Mamba2 recurrent cell, last-timestep output only.
// Key simplification: dA_t is scalar per (b,h), so
//   y_T = sum_t (prod_{s>t} dA_s) * dt_t * (B_t . C_T) * x_t  + D * x_T
// which is a forward recurrence on a 64-vector per (b,h). No 64x64 state.
//
// Pipeline (all on `stream`):
//  1) gemm_wmma   : P = seq @ W_in[:,2048:4256]   (4096x1024 @ 1024x2208, fp32 WMMA,
//                   double-buffered LDS fed by GLOBAL_LOAD_ASYNC_TO_LDS)
//  2) ct_kernel   : C_T (conv+silu at t=T-1, C channels)
//  3) s_kernel    : s[b,t] = conv_silu(B_t) . C_T
//  4) scal_kernel : dt[b,t,h]=softplus(..), dA[b,t,h]=exp(dt*A)
//  5) chunk_kernel: per (b,h,chunk of 64 steps): affine map (W, y)
//  6) combine     : compose chunks, add D*x_T  -> yraw (4x2048)
//  7) z_kernel    : z_T = x_t @ W_in[:,0:2048]  (GEMV)
//  8) gate_kernel : g = yraw*silu(z); invrms per b; g *= norm_w
//  9) out_kernel  : out = (g*invrms) @ W_out    (GEMV)
// ---------------------------------------------------------------------------

#define D_MODEL   1024
#define D_STATE   64
#define D_CONV    4
#define D_INNER   2048
#define NHEADS    32
#define HEADDIM   64
#define CONV_DIM  2176           // D_INNER + 2*D_STATE
#define D_IN_PROJ 4256
#define B_SZ      4
#define T_SEQ     1024           // T_PREV + 1
#define M_ROWS    (B_SZ * T_SEQ) // 4096
#define N_P       2208           // CONV_DIM + NHEADS (W_in cols 2048..4255)
#define NCHUNK    16
#define CHLEN     64             // T_SEQ / NCHUNK

typedef __attribute__((ext_vector_type(2))) float v2f;
typedef __attribute__((ext_vector_type(8))) float v8f;
typedef int v4i __attribute__((vector_size(16)));   // matches builtin param type

#define GAS __attribute__((address_space(1)))       // global
#define LAS __attribute__((address_space(3)))       // LDS

#if defined(__has_builtin)
#if __has_builtin(__builtin_amdgcn_global_load_async_to_lds_b128) && \
    __has_builtin(__builtin_amdgcn_global_load_async_to_lds_b32)
#define HAVE_ASYNC_LDS 1
#endif
#endif

#if defined(__has_builtin)
#if __has_builtin(__builtin_amdgcn_s_wait_asynccnt)
#define WAIT_ASYNC0() __builtin_amdgcn_s_wait_asynccnt(0)
#else
#define WAIT_ASYNC0() asm volatile("s_wait_asynccnt 0x0" ::: "memory")
#endif
#else
#define WAIT_ASYNC0() asm volatile("s_wait_asynccnt 0x0" ::: "memory")
#endif

__device__ __forceinline__ float siluf(float x)     { return x / (1.0f + __expf(-x)); }
__device__ __forceinline__ float softplusf(float x) { return (x > 20.0f) ? x : log1pf(__expf(x)); }

// Row r of the logical sequence (b = r/1024, t = r%1024): history for t<1023, x_t at t=1023.
__device__ __forceinline__ const float* seq_row(const float* hist, const float* xt, int r) {
    int b = r >> 10, t = r & 1023;
    return (t < 1023) ? (hist + (size_t)(b * 1023 + t) * D_MODEL)
                      : (xt   + (size_t)b * D_MODEL);
}

// ---------------------------------------------------------------------------
// 1) Main GEMM: P[4096][2208] = seq @ W_in[:, 2048:4256] with f32 WMMA 16x16x4.
//    Block tile 128x128, K-slab 32 in LDS (double-buffered). 8 waves:
//    (wr 0..3)x(wc 0..1), wave tile = 32 rows x 64 cols = 2x4 16x16 accs.
// A-frag (16x4 f32, M x K): lanes 0-15 hold {K=0,K=1}, lanes 16-31 hold {K=2,K=3},
// M = lane%16 -> b64 load from As[row][k + (lane>>4)*2].
// B-frag (4x16, K x N) stored N-major in LDS -> identical addressing.
// B tile is TRANSPOSED on the fly by GLOBAL_LOAD_ASYNC_TO_LDS_B32 (per-lane
// LDS destination addresses); A tile streams with ASYNC_TO_LDS_B128.
// C/D 16x16 f32: VGPR v holds M=v (lanes 0-15) / M=v+8 (lanes 16-31), N=lane%16.
// ---------------------------------------------------------------------------
#define BM 128
#define BN 128
#define BKK 32
#define LDP 36   // LDS row pitch (floats): keeps b64/b128 alignment, spreads banks

__global__ __launch_bounds__(256)
void mamba_gemm_wmma(const float* __restrict__ hist, const float* __restrict__ xt,
                     const float* __restrict__ Win, float* __restrict__ P)
{
#ifdef HAVE_ASYNC_LDS
    __shared__ float As[2][BM * LDP];
    __shared__ float Bs[2][BN * LDP];
#else
    __shared__ float As[1][BM * LDP];
    __shared__ float Bs[1][BN * LDP];
#endif

    const int tid   = threadIdx.x;
    const int lane  = tid & 31;
    const int wave  = tid >> 5;
    const int wr    = wave >> 1;           // 0..3 : 32-row band
    const int wc    = wave & 1;            // 0..1 : 64-col half
    const int mBase = blockIdx.x * BM;     // 32 blocks
    const int nBase = blockIdx.y * BN;     // 18 blocks (last partially OOB in N)

    v8f acc[2][4];
#pragma unroll
    for (int i = 0; i < 2; ++i)
#pragma unroll
        for (int j = 0; j < 4; ++j) acc[i][j] = (v8f)0.0f;

    const int arow = tid >> 3;             // 0..31 (+rep*32)
    const int acol = (tid & 7) * 4;
    const int brow = tid >> 5;             // 0..7  (+rep*8)
    const int bcol = (tid & 31) * 4;

    const int l16  = lane & 15;
    const int khav = (lane >> 4) * 2;      // 0 or 2

#ifdef HAVE_ASYNC_LDS
    // Issue the async DMA of one K-slab into LDS buffer `buf`.
    auto issue = [&](int buf, int kb) {
#pragma unroll
        for (int rep = 0; rep < 4; ++rep) {           // A: 128x32, b128 per lane
            int r = arow + rep * 32;
            const float* src = seq_row(hist, xt, mBase + r) + kb + acol;
            __builtin_amdgcn_global_load_async_to_lds_b128(
                (GAS v4i*)(uintptr_t)src,
                (LAS v4i*)(uint32_t)(uintptr_t)&As[buf][r * LDP + acol], 0, 0);
        }
#pragma unroll
        for (int rep = 0; rep < 4; ++rep) {           // B: 32x128, transposed b32
            int r = brow + rep * 8;                   // k within slab
            const float* src = Win + (size_t)(kb + r) * D_IN_PROJ + (size_t)(D_INNER + nBase + bcol);
#pragma unroll
            for (int i = 0; i < 4; ++i) {
                if (nBase + bcol + i < N_P) {         // EXEC-guarded; dead cols feed
                    __builtin_amdgcn_global_load_async_to_lds_b32(   // only discarded accs
                        (GAS int*)(uintptr_t)(src + i),
                        (LAS int*)(uint32_t)(uintptr_t)&Bs[buf][(bcol + i) * LDP + r], 0, 0);
                }
            }
        }
    };

    issue(0, 0);
    WAIT_ASYNC0();
    __syncthreads();

    for (int kb = 0; kb < D_MODEL; kb += BKK) {
        const int cur = (kb / BKK) & 1;
        if (kb + BKK < D_MODEL) issue(cur ^ 1, kb + BKK);   // prefetch next slab

        const float* Asb = &As[cur][0];
        const float* Bsb = &Bs[cur][0];
#pragma unroll
        for (int kk = 0; kk < BKK; kk += 4) {
            const int ko = kk + khav;
            v2f a0 = *(const v2f*)&Asb[(wr * 32 +  0 + l16) * LDP + ko];
            v2f a1 = *(const v2f*)&Asb[(wr * 32 + 16 + l16) * LDP + ko];
            v2f b0 = *(const v2f*)&Bsb[(wc * 64 +  0 + l16) * LDP + ko];
            v2f b1 = *(const v2f*)&Bsb[(wc * 64 + 16 + l16) * LDP + ko];
            v2f b2 = *(const v2f*)&Bsb[(wc * 64 + 32 + l16) * LDP + ko];
            v2f b3 = *(const v2f*)&Bsb[(wc * 64 + 48 + l16) * LDP + ko];
            acc[0][0] = __builtin_amdgcn_wmma_f32_16x16x4_f32(false, a0, false, b0, (short)0, acc[0][0], false, false);
            acc[0][1] = __builtin_amdgcn_wmma_f32_16x16x4_f32(false, a0, false, b1, (short)0, acc[0][1], false, false);
            acc[0][2] = __builtin_amdgcn_wmma_f32_16x16x4_f32(false, a0, false, b2, (short)0, acc[0][2], false, false);
            acc[0][3] = __builtin_amdgcn_wmma_f32_16x16x4_f32(false, a0, false, b3, (short)0, acc[0][3], false, false);
            acc[1][0] = __builtin_amdgcn_wmma_f32_16x16x4_f32(false, a1, false, b0, (short)0, acc[1][0], false, false);
            acc[1][1] = __builtin_amdgcn_wmma_f32_16x16x4_f32(false, a1, false, b1, (short)0, acc[1][1], false, false);
            acc[1][2] = __builtin_amdgcn_wmma_f32_16x16x4_f32(false, a1, false, b2, (short)0, acc[1][2], false, false);
            acc[1][3] = __builtin_amdgcn_wmma_f32_16x16x4_f32(false, a1, false, b3, (short)0, acc[1][3], false, false);
        }
        WAIT_ASYNC0();       // next-slab DMA done (wave-local) ...
        __syncthreads();     // ... and for every wave in the group
    }
#else
    // Fallback: synchronous staging through VGPRs.
    for (int kb = 0; kb < D_MODEL; kb += BKK) {
#pragma unroll
        for (int rep = 0; rep < 4; ++rep) {
            int r = arow + rep * 32;
            const float* src = seq_row(hist, xt, mBase + r) + kb + acol;
            const float4 v = *(const float4*)src;
            *(float4*)&As[0][r * LDP + acol] = v;
        }
#pragma unroll
        for (int rep = 0; rep < 4; ++rep) {
            int r = brow + rep * 8;
            const float* src = Win + (size_t)(kb + r) * D_IN_PROJ + (size_t)(D_INNER + nBase + bcol);
#pragma unroll
            for (int i = 0; i < 4; ++i) {
                float v = (nBase + bcol + i < N_P) ? src[i] : 0.0f;
                Bs[0][(bcol + i) * LDP + r] = v;
            }
        }
        __syncthreads();
#pragma unroll
        for (int kk = 0; kk < BKK; kk += 4) {
            const int ko = kk + khav;
            v2f a0 = *(const v2f*)&As[0][(wr * 32 +  0 + l16) * LDP + ko];
            v2f a1 = *(const v2f*)&As[0][(wr * 32 + 16 + l16) * LDP + ko];
            v2f b0 = *(const v2f*)&Bs[0][(wc * 64 +  0 + l16) * LDP + ko];
            v2f b1 = *(const v2f*)&Bs[0][(wc * 64 + 16 + l16) * LDP + ko];
            v2f b2 = *(const v2f*)&Bs[0][(wc * 64 + 32 + l16) * LDP + ko];
            v2f b3 = *(const v2f*)&Bs[0][(wc * 64 + 48 + l16) * LDP + ko];
            acc[0][0] = __builtin_amdgcn_wmma_f32_16x16x4_f32(false, a0, false, b0, (short)0, acc[0][0], false, false);
            acc[0][1] = __builtin_amdgcn_wmma_f32_16x16x4_f32(false, a0, false, b1, (short)0, acc[0][1], false, false);
            acc[0][2] = __builtin_amdgcn_wmma_f32_16x16x4_f32(false, a0, false, b2, (short)0, acc[0][2], false, false);
            acc[0][3] = __builtin_amdgcn_wmma_f32_16x16x4_f32(false, a0, false, b3, (short)0, acc[0][3], false, false);
            acc[1][0] = __builtin_amdgcn_wmma_f32_16x16x4_f32(false, a1, false, b0, (short)0, acc[1][0], false, false);
            acc[1][1] = __builtin_amdgcn_wmma_f32_16x16x4_f32(false, a1, false, b1, (short)0, acc[1][1], false, false);
            acc[1][2] = __builtin_amdgcn_wmma_f32_16x16x4_f32(false, a1, false, b2, (short)0, acc[1][2], false, false);
            acc[1][3] = __builtin_amdgcn_wmma_f32_16x16x4_f32(false, a1, false, b3, (short)0, acc[1][3], false, false);
        }
        __syncthreads();
    }
#endif

    // ---- store D tiles ----
    const int rAdd = (lane >> 4) * 8;      // +8 for lanes 16..31
#pragma unroll
    for (int mt = 0; mt < 2; ++mt) {
#pragma unroll
        for (int nt = 0; nt < 4; ++nt) {
            int col = nBase + wc * 64 + nt * 16 + l16;
            if (col < N_P) {
#pragma unroll
                for (int v = 0; v < 8; ++v) {
                    int row = mBase + wr * 32 + mt * 16 + v + rAdd;
                    P[(size_t)row * N_P + col] = acc[mt][nt][v];
                }
            }
        }
    }
}

// ---------------------------------------------------------------------------
// 2) C_T : conv+silu of C channels (n = 2112..2175) at t = T-1. 1 block x 256.
// ---------------------------------------------------------------------------
__global__ void mamba_ct(const float* __restrict__ P, const float* __restrict__ cw,
                         const float* __restrict__ cb, float* __restrict__ CT)
{
    int tid = threadIdx.x;
    int b = tid >> 6, j = tid & 63;
    int c = D_INNER + D_STATE + j;               // xBC channel index == P column
    float acc = cb[c];
#pragma unroll
    for (int k = 0; k < D_CONV; ++k) {
        int t = (T_SEQ - 1) - 3 + k;
        acc += cw[c * D_CONV + k] * P[((size_t)b * T_SEQ + t) * N_P + c];
    }
    CT[b * 64 + j] = siluf(acc);
}

// ---------------------------------------------------------------------------
// 3) s[b,t] = conv_silu(B_t) . C_T.  grid (256, 4) x 256: 4 timesteps / block.
// ---------------------------------------------------------------------------
__global__ void mamba_s(const float* __restrict__ P, const float* __restrict__ cw,
                        const float* __restrict__ cb, const float* __restrict__ CT,
                        float* __restrict__ sArr)
{
    __shared__ float red[8];
    int tid = threadIdx.x;
    int tl = tid >> 6, j = tid & 63;
    int t = blockIdx.x * 4 + tl;
    int b = blockIdx.y;
    int c = D_INNER + j;                          // B channels: n = 2048..2111
    float acc = cb[c];
#pragma unroll
    for (int k = 0; k < D_CONV; ++k) {
        int tt = t - 3 + k;
        if (tt >= 0) acc += cw[c * D_CONV + k] * P[((size_t)b * T_SEQ + tt) * N_P + c];
    }
    float v = siluf(acc) * CT[b * 64 + j];
#pragma unroll
    for (int off = 16; off > 0; off >>= 1) v += __shfl_down(v, off, 32);
    if ((tid & 31) == 0) red[tid >> 5] = v;
    __syncthreads();
    if ((tid & 63) == 0) sArr[b * T_SEQ + t] = red[tid >> 5] + red[(tid >> 5) + 1];
}

// ---------------------------------------------------------------------------
// 4) dt[b,t,h] = softplus(P_dt + bias); dA = exp(dt * -exp(A_log)). 512 x 256.
// ---------------------------------------------------------------------------
__global__ void mamba_scal(const float* __restrict__ P, const float* __restrict__ dt_bias,
                           const float* __restrict__ A_log,
                           float* __restrict__ dtv, float* __restrict__ dAv)
{
    int idx = blockIdx.x * 256 + threadIdx.x;     // (b*1024 + t)*32 + h
    int h = idx & 31;
    int bt = idx >> 5;                            // b*1024 + t
    float raw = P[(size_t)bt * N_P + CONV_DIM + h] + dt_bias[h];
    float dt = softplusf(raw);
    float A  = -__expf(A_log[h]);
    dtv[idx] = dt;
    dAv[idx] = __expf(dt * A);
}

// ---------------------------------------------------------------------------
// 5) Chunked recurrence: each block = (b, h, chunk of 64 steps), 64 threads.
//    Produces affine map (W = prod dA, y = chunk contribution from h_in=0).
// ---------------------------------------------------------------------------
__global__ __launch_bounds__(64)
void mamba_chunk(const float* __restrict__ P, const float* __restrict__ cw,
                 const float* __restrict__ cb, const float* __restrict__ dtv,
                 const float* __restrict__ dAv, const float* __restrict__ sArr,
                 float* __restrict__ ycs, float* __restrict__ Wc,
                 float* __restrict__ xlast)
{
    int bid = blockIdx.x;
    int chunk = bid & (NCHUNK - 1);
    int h = (bid >> 4) & (NHEADS - 1);
    int b = bid >> 9;
    int p = threadIdx.x;
    int c = h * HEADDIM + p;                      // x channel == P column
    int t0 = chunk * CHLEN;

    const float w0 = cw[c * D_CONV + 0], w1 = cw[c * D_CONV + 1];
    const float w2 = cw[c * D_CONV + 2], w3 = cw[c * D_CONV + 3];
    const float bc = cb[c];

    // rolling conv window: q1..q3 = P at t0-3..t0-1 (zero-padded)
    float q1 = (t0 - 3 >= 0) ? P[((size_t)b * T_SEQ + t0 - 3) * N_P + c] : 0.0f;
    float q2 = (t0 - 2 >= 0) ? P[((size_t)b * T_SEQ + t0 - 2) * N_P + c] : 0.0f;
    float q3 = (t0 - 1 >= 0) ? P[((size_t)b * T_SEQ + t0 - 1) * N_P + c] : 0.0f;

    float y = 0.0f, W = 1.0f, x = 0.0f;
    for (int i = 0; i < CHLEN; ++i) {
        int t = t0 + i;
        float q0 = q1; q1 = q2; q2 = q3;
        q3 = P[((size_t)b * T_SEQ + t) * N_P + c];
        x = siluf(bc + w0 * q0 + w1 * q1 + w2 * q2 + w3 * q3);
        int bt = b * T_SEQ + t;
        float dt = dtv[bt * 32 + h];
        float dA = dAv[bt * 32 + h];
        float coef = dt * sArr[bt];
        y = dA * y + coef * x;
        W *= dA;
    }
    int bh = b * NHEADS + h;
    ycs[((size_t)bh * NCHUNK + chunk) * HEADDIM + p] = y;
    if (p == 0) Wc[bh * NCHUNK + chunk] = W;
    if (chunk == NCHUNK - 1) xlast[bh * HEADDIM + p] = x;
}

// ---------------------------------------------------------------------------
// 6) Compose chunks in time order, add D * x_T. 128 blocks x 64.
// ---------------------------------------------------------------------------
__global__ __launch_bounds__(64)
void mamba_combine(const float* __restrict__ ycs, const float* __restrict__ Wc,
                   const float* __restrict__ xlast, const float* __restrict__ Dp,
                   float* __restrict__ yraw)
{
    int bh = blockIdx.x;                          // b*32 + h
    int h = bh & (NHEADS - 1);
    int b = bh >> 5;
    int p = threadIdx.x;
    float y = 0.0f;
#pragma unroll
    for (int j = 0; j < NCHUNK; ++j)
        y = Wc[bh * NCHUNK + j] * y + ycs[((size_t)bh * NCHUNK + j) * HEADDIM + p];
    y += Dp[h] * xlast[bh * HEADDIM + p];
    yraw[b * D_INNER + h * HEADDIM + p] = y;
}

// ---------------------------------------------------------------------------
// 7) z_T = x_t @ W_in[:, 0:2048]. 32 blocks x 64; x_t cached in LDS,
//    4 batch accumulators per thread so W_in is read once.
// ---------------------------------------------------------------------------
__global__ __launch_bounds__(64)
void mamba_z(const float* __restrict__ xt, const float* __restrict__ Win,
             float* __restrict__ zbuf)
{
    __shared__ float xs[B_SZ * D_MODEL];
    int tid = threadIdx.x;
    for (int i = tid; i < B_SZ * D_MODEL; i += 64) xs[i] = xt[i];
    __syncthreads();
    int col = blockIdx.x * 64 + tid;
    float acc0 = 0, acc1 = 0, acc2 = 0, acc3 = 0;
    for (int k = 0; k < D_MODEL; ++k) {
        float w = Win[(size_t)k * D_IN_PROJ + col];
        acc0 += xs[0 * D_MODEL + k] * w;
        acc1 += xs[1 * D_MODEL + k] * w;
        acc2 += xs[2 * D_MODEL + k] * w;
        acc3 += xs[3 * D_MODEL + k] * w;
    }
    zbuf[0 * D_INNER + col] = acc0;
    zbuf[1 * D_INNER + col] = acc1;
    zbuf[2 * D_INNER + col] = acc2;
    zbuf[3 * D_INNER + col] = acc3;
}

// ---------------------------------------------------------------------------
// 8) Gate + RMS stats: g = yraw*silu(z); invrms[b]; g *= norm_w. 4 blocks x 256.
// ---------------------------------------------------------------------------
__global__ void mamba_gate(const float* __restrict__ yraw, const float* __restrict__ zbuf,
                           const float* __restrict__ nw, float* __restrict__ gbuf,
                           float* __restrict__ invrms)
{
    __shared__ float red[256];
    int b = blockIdx.x, tid = threadIdx.x;
    float ss = 0.0f;
#pragma unroll
    for (int i = 0; i < D_INNER / 256; ++i) {
        int d = tid + i * 256;
        float g = yraw[b * D_INNER + d] * siluf(zbuf[b * D_INNER + d]);
        ss += g * g;
        gbuf[b * D_INNER + d] = g * nw[d];
    }
    red[tid] = ss;
    __syncthreads();
    for (int off = 128; off > 0; off >>= 1) {
        if (tid < off) red[tid] += red[tid + off];
        __syncthreads();
    }
    if (tid == 0) invrms[b] = rsqrtf(red[0] / (float)D_INNER + 1e-5f);
}

// ---------------------------------------------------------------------------
// 9) out = (g * invrms) @ W_out. 16 blocks x 64; g cached in LDS.
// ---------------------------------------------------------------------------
__global__ __launch_bounds__(64)
void mamba_out(const float* __restrict__ gbuf, const float* __restrict__ invrms,
               const float* __restrict__ Wout, float* __restrict__ out)
{
    __shared__ float gs[B_SZ * D_INNER];
    int tid = threadIdx.x;
    for (int i = tid; i < B_SZ * D_INNER; i += 64) gs[i] = gbuf[i];
    __syncthreads();
    int m = blockIdx.x * 64 + tid;
    float acc0 = 0, acc1 = 0, acc2 = 0, acc3 = 0;
    for (int d = 0; d < D_INNER; ++d) {
        float w = Wout[(size_t)d * D_MODEL + m];
        acc0 += gs[0 * D_INNER + d] * w;
        acc1 += gs[1 * D_INNER + d] * w;
        acc2 += gs[2 * D_INNER + d] * w;
        acc3 += gs[3 * D_INNER + d] * w;
    }
    out[0 * D_MODEL + m] = acc0 * invrms[0];
    out[1 * D_MODEL + m] = acc1 * invrms[1];
    out[2 * D_MODEL + m] = acc2 * invrms[2];
    out[3 * D_MODEL + m] = acc3 * invrms[3];
}

// ---------------------------------------------------------------------------
// Launch
// ---------------------------------------------------------------------------
extern "C" void kernel_launch(void* const* d_in, const int* in_sizes, int n_in,
                              void* d_out, int out_size, void* d_ws, size_t ws_size,
                              hipStream_t stream) {
    const float* x_t     = (const float*)d_in[0];
    const float* history = (const float*)d_in[1];
    const float* W_in    = (const float*)d_in[2];
    const float* conv_w  = (const float*)d_in[3];
    const float* conv_b  = (const float*)d_in[4];
    const float* dt_bias = (const float*)d_in[5];
    const float* A_log   = (const float*)d_in[6];
    const float* D_param = (const float*)d_in[7];
    const float* norm_w  = (const float*)d_in[8];
    const float* W_out   = (const float*)d_in[9];
    float* out = (float*)d_out;

    float* ws = (float*)d_ws;
    // workspace layout (floats), ~37.9 MB total
    float* P      = ws;                                    // 4096*2208
    float* CT     = P      + (size_t)M_ROWS * N_P;         // 256
    float* sArr   = CT     + 256;                          // 4096
    float* dtv    = sArr   + 4096;                         // 131072
    float* dAv    = dtv    + 131072;                       // 131072
    float* ycs    = dAv    + 131072;                       // 131072
    float* Wc     = ycs    + 131072;                       // 2048
    float* xlast  = Wc     + 2048;                         // 8192
    float* yraw   = xlast  + 8192;                         // 8192
    float* zbuf   = yraw   + 8192;                         // 8192
    float* gbuf   = zbuf   + 8192;                         // 8192
    float* invrms = gbuf   + 8192;                         // 4

    mamba_gemm_wmma<<<dim3(M_ROWS / BM, (N_P + BN - 1) / BN), 256, 0, stream>>>(history, x_t, W_in, P);
    mamba_ct<<<1, 256, 0, stream>>>(P, conv_w, conv_b, CT);
    mamba_s<<<dim3(T_SEQ / 4, B_SZ), 256, 0, stream>>>(P, conv_w, conv_b, CT, sArr);
    mamba_scal<<<(B_SZ * T_SEQ * NHEADS) / 256, 256, 0, stream>>>(P, dt_bias, A_log, dtv, dAv);
    mamba_chunk<<<B_SZ * NHEADS * NCHUNK, 64, 0, stream>>>(P, conv_w, conv_b, dtv, dAv, sArr, ycs, Wc, xlast);
    mamba_combine<<<B_SZ * NHEADS, 64, 0, stream>>>(ycs, Wc, xlast, D_param, yraw);
    mamba_z<<<D_INNER / 64, 64, 0, stream>>>(x_t, W_in, zbuf);
    mamba_gate<<<B_SZ, 256, 0, stream>>>(yraw, zbuf, norm_w, gbuf, invrms);
    mamba_out<<<D_MODEL / 64, 64, 0, stream>>>(gbuf, invrms, W_out, out);
}